// FastAttention_6640019439753
// MI455X (gfx1250) — compile-verified
//
#include <hip/hip_runtime.h>
#include <stdint.h>

typedef float v2f __attribute__((ext_vector_type(2)));
typedef float v8f __attribute__((ext_vector_type(8)));

#define TPB    256
#define N_CTX  2048
#define D_DIM  64
#define M_DIM  256
#define NT_A   128
#define NT_B   64
#define PSTR   68    // padded row stride (floats) for proj/k/v/q tiles (272B, 16B aligned)
#define KT_STR 130   // k'^T staging stride
#define QP_STR 258   // q' staging stride
#define CTX_STR 65   // context stride

// LDS layout (float offsets)
#define OFF_PROJ  0                    // 256*68  = 17408
#define OFF_K     17408                // 128*68  = 8704
#define OFF_V     26112                // 128*68  = 8704
#define OFF_CTX   17408                // 256*65  = 16640 (reuses K+V region in phase B)
#define OFF_STAGE 34816                // 256*130 = 33280 (phase A: k'^T)
#define OFF_QT    34816                // 64*68   = 4352  (phase B: q tile)
#define OFF_QP    39168                // 64*258  = 16512 (phase B: q')
#define OFF_KSUM  68096                // 256
#define OFF_DIAG  68352                // 128
#define OFF_DROW  68480                // 64
#define OFF_STAB  68544                // 2*64
#define SMEM_FLOATS 68672
#define SMEM_BYTES  (SMEM_FLOATS * 4)

__device__ __forceinline__ v8f zero8() {
    v8f z = {0.f, 0.f, 0.f, 0.f, 0.f, 0.f, 0.f, 0.f};
    return z;
}

__device__ __forceinline__ v8f wmma4(v2f a, v2f b, v8f c) {
    // V_WMMA_F32_16X16X4_F32 : D = A(16x4,f32) x B(4x16,f32) + C(16x16,f32)
    return __builtin_amdgcn_wmma_f32_16x16x4_f32(false, a, false, b, (short)0, c,
                                                 false, false);
}

__device__ __forceinline__ void async_cp16(float* smem, uint32_t foff, const float* g) {
    uint32_t lds_byte = (uint32_t)(uintptr_t)(smem + foff);  // flat low 32b == LDS offset
    asm volatile("global_load_async_to_lds_b128 %0, %1, off"
                 :: "v"(lds_byte), "v"(g) : "memory");
}
__device__ __forceinline__ void wait_async() {
    asm volatile("s_wait_asynccnt 0" ::: "memory");
}

__global__ __launch_bounds__(TPB, 1)
void performer_attn_kernel(const float* __restrict__ q, const float* __restrict__ k,
                           const float* __restrict__ v, const float* __restrict__ proj,
                           float* __restrict__ out) {
    extern __shared__ float smem[];
    const int tid  = threadIdx.x;
    const int lane = tid & 31;
    const int w    = tid >> 5;       // wave id 0..7
    const int l15  = lane & 15;
    const int hi   = lane >> 4;      // half-wave: 0 or 1

    const size_t base = (size_t)blockIdx.x * N_CTX * D_DIM;
    const float* qg = q + base;
    const float* kg = k + base;
    const float* vg = v + base;
    float*       og = out + base;

    const float DN    = 0.35355339059327379f; // 64^-0.25
    const float RATIO = 0.0625f;              // 256^-0.5
    const float DIAGC = 0.0625f;              // 0.5 * DN^2
    const float EPS   = 1e-4f;

    // ---- stage projection matrix [256x64] -> [256][PSTR] ----
    for (int i = tid; i < M_DIM * (D_DIM / 4); i += TPB) {
        int r = i >> 4, c4 = i & 15;
        async_cp16(smem, OFF_PROJ + r * PSTR + c4 * 4, proj + r * D_DIM + c4 * 4);
    }
    smem[OFF_KSUM + tid] = 0.0f;    // TPB == M_DIM
    wait_async();
    __syncthreads();

    // ================= Phase A: context = k'^T v, ksum = sum_n k' =================
    v8f ctx[2][4];
    for (int i = 0; i < 2; i++)
        for (int e = 0; e < 4; e++) ctx[i][e] = zero8();

    for (int n0 = 0; n0 < N_CTX; n0 += NT_A) {
        for (int i = tid; i < NT_A * (D_DIM / 4); i += TPB) {
            int r = i >> 4, c4 = i & 15;
            async_cp16(smem, OFF_K + r * PSTR + c4 * 4, kg + (size_t)(n0 + r) * D_DIM + c4 * 4);
            async_cp16(smem, OFF_V + r * PSTR + c4 * 4, vg + (size_t)(n0 + r) * D_DIM + c4 * 4);
        }
        if (n0 + NT_A < N_CTX) __builtin_prefetch(kg + (size_t)(n0 + NT_A) * D_DIM, 0, 0);
        wait_async();
        __syncthreads();

        if (tid < NT_A) {   // diag(k) rows
            float s = 0.f;
            const float* row = &smem[OFF_K + tid * PSTR];
            #pragma unroll
            for (int dd = 0; dd < D_DIM; dd++) { float x = row[dd]; s += x * x; }
            smem[OFF_DIAG + tid] = s * DIAGC;
        }
        __syncthreads();

        // ---- k_dash GEMM: wave w owns rows [w*16, w*16+16), all 16 m-tiles (2 groups) ----
        for (int g = 0; g < 2; g++) {
            v8f acc[8];
            #pragma unroll
            for (int mi = 0; mi < 8; mi++) acc[mi] = zero8();
            for (int kc = 0; kc < 16; kc++) {
                v2f a = *(const v2f*)&smem[OFF_K + (w * 16 + l15) * PSTR + kc * 4 + hi * 2];
                #pragma unroll
                for (int mi = 0; mi < 8; mi++) {
                    int mt = g * 8 + mi;
                    v2f b = *(const v2f*)&smem[OFF_PROJ + (mt * 16 + l15) * PSTR + kc * 4 + hi * 2];
                    acc[mi] = wmma4(a, b, acc[mi]);
                }
            }
            // epilogue: k' = ratio*exp(DN*dash - diag + eps); write k'^T; ksum
            #pragma unroll
            for (int mi = 0; mi < 8; mi++) {
                int m = (g * 8 + mi) * 16 + l15;
                float colsum = 0.f;
                #pragma unroll
                for (int j = 0; j < 8; j++) {
                    int rloc = w * 16 + j + hi * 8;
                    float dash = DN * acc[mi][j];
                    float kp = RATIO * __expf(dash - smem[OFF_DIAG + rloc] + EPS);
                    smem[OFF_STAGE + m * KT_STR + rloc] = kp;
                    colsum += kp;
                }
                colsum += __shfl_xor(colsum, 16, 32);
                if (hi == 0) atomicAdd(&smem[OFF_KSUM + m], colsum);
            }
        }
        __syncthreads();

        // ---- context accumulation: wave w owns m-tiles {2w,2w+1} x e-tiles 0..3 ----
        for (int kc = 0; kc < NT_A / 4; kc++) {
            v2f a0 = *(const v2f*)&smem[OFF_STAGE + ((2 * w + 0) * 16 + l15) * KT_STR + kc * 4 + hi * 2];
            v2f a1 = *(const v2f*)&smem[OFF_STAGE + ((2 * w + 1) * 16 + l15) * KT_STR + kc * 4 + hi * 2];
            int kn = kc * 4 + hi * 2;
            #pragma unroll
            for (int et = 0; et < 4; et++) {
                v2f b;
                b.x = smem[OFF_V + kn * PSTR + et * 16 + l15];
                b.y = smem[OFF_V + (kn + 1) * PSTR + et * 16 + l15];
                ctx[0][et] = wmma4(a0, b, ctx[0][et]);
                ctx[1][et] = wmma4(a1, b, ctx[1][et]);
            }
        }
        __syncthreads();
    }

    // ---- context registers -> LDS [m][CTX_STR] ----
    #pragma unroll
    for (int i = 0; i < 2; i++) {
        int mt = 2 * w + i;
        #pragma unroll
        for (int et = 0; et < 4; et++) {
            #pragma unroll
            for (int j = 0; j < 8; j++) {
                int m = mt * 16 + j + hi * 8;
                smem[OFF_CTX + m * CTX_STR + et * 16 + l15] = ctx[i][et][j];
            }
        }
    }
    __syncthreads();

    // ================= Phase B: out = (q' @ context) * Dinv =================
    const int nb = w >> 1;   // row-block of 16 within the 64-row tile
    for (int n0 = 0; n0 < N_CTX; n0 += NT_B) {
        for (int i = tid; i < NT_B * (D_DIM / 4); i += TPB) {
            int r = i >> 4, c4 = i & 15;
            async_cp16(smem, OFF_QT + r * PSTR + c4 * 4, qg + (size_t)(n0 + r) * D_DIM + c4 * 4);
        }
        if (n0 + NT_B < N_CTX) __builtin_prefetch(qg + (size_t)(n0 + NT_B) * D_DIM, 0, 0);
        wait_async();
        __syncthreads();

        if (tid < NT_B) {
            float s = 0.f;
            const float* row = &smem[OFF_QT + tid * PSTR];
            #pragma unroll
            for (int dd = 0; dd < D_DIM; dd++) { float x = row[dd]; s += x * x; }
            smem[OFF_DIAG + tid] = s * DIAGC;
            smem[OFF_DROW + tid] = 0.f;
        }
        __syncthreads();

        // ---- q_dash GEMM: wave w owns row-block nb, m-tiles (w&1)*8 .. +7 ----
        v8f qd[8];
        #pragma unroll
        for (int mi = 0; mi < 8; mi++) qd[mi] = zero8();
        for (int kc = 0; kc < 16; kc++) {
            v2f a = *(const v2f*)&smem[OFF_QT + (nb * 16 + l15) * PSTR + kc * 4 + hi * 2];
            #pragma unroll
            for (int mi = 0; mi < 8; mi++) {
                int mt = (w & 1) * 8 + mi;
                v2f b = *(const v2f*)&smem[OFF_PROJ + (mt * 16 + l15) * PSTR + kc * 4 + hi * 2];
                qd[mi] = wmma4(a, b, qd[mi]);
            }
        }
        // scale to dash, partial row-max -> LDS
        #pragma unroll
        for (int j = 0; j < 8; j++) {
            float mx = -3.402823466e+38f;
            #pragma unroll
            for (int mi = 0; mi < 8; mi++) {
                qd[mi][j] *= DN;
                mx = fmaxf(mx, qd[mi][j]);
            }
            mx = fmaxf(mx, __shfl_xor(mx, 1, 32));
            mx = fmaxf(mx, __shfl_xor(mx, 2, 32));
            mx = fmaxf(mx, __shfl_xor(mx, 4, 32));
            mx = fmaxf(mx, __shfl_xor(mx, 8, 32));
            if (l15 == 0) smem[OFF_STAB + (w & 1) * 64 + nb * 16 + j + hi * 8] = mx;
        }
        __syncthreads();

        // q' epilogue + D row-sums + stage q'
        float dpart[8] = {0.f, 0.f, 0.f, 0.f, 0.f, 0.f, 0.f, 0.f};
        #pragma unroll
        for (int mi = 0; mi < 8; mi++) {
            int m = ((w & 1) * 8 + mi) * 16 + l15;
            float ks = smem[OFF_KSUM + m];
            #pragma unroll
            for (int j = 0; j < 8; j++) {
                int rloc = nb * 16 + j + hi * 8;
                float stab = fmaxf(smem[OFF_STAB + rloc], smem[OFF_STAB + 64 + rloc]);
                float qp = RATIO * (__expf(qd[mi][j] - smem[OFF_DIAG + rloc] - stab) + EPS);
                smem[OFF_QP + rloc * QP_STR + m] = qp;
                dpart[j] += qp * ks;
            }
        }
        #pragma unroll
        for (int j = 0; j < 8; j++) {
            float s = dpart[j];
            s += __shfl_xor(s, 1, 32);
            s += __shfl_xor(s, 2, 32);
            s += __shfl_xor(s, 4, 32);
            s += __shfl_xor(s, 8, 32);
            if (l15 == 0) atomicAdd(&smem[OFF_DROW + nb * 16 + j + hi * 8], s);
        }
        __syncthreads();

        // ---- out GEMM: wave w owns row-block nb, e-tiles (w&1)*2 .. +1 ----
        v8f o[2];
        o[0] = zero8();
        o[1] = zero8();
        for (int kc = 0; kc < M_DIM / 4; kc++) {
            v2f a = *(const v2f*)&smem[OFF_QP + (nb * 16 + l15) * QP_STR + kc * 4 + hi * 2];
            int km = kc * 4 + hi * 2;
            #pragma unroll
            for (int i2 = 0; i2 < 2; i2++) {
                int et = (w & 1) * 2 + i2;
                v2f b;
                b.x = smem[OFF_CTX + km * CTX_STR + et * 16 + l15];
                b.y = smem[OFF_CTX + (km + 1) * CTX_STR + et * 16 + l15];
                o[i2] = wmma4(a, b, o[i2]);
            }
        }
        #pragma unroll
        for (int i2 = 0; i2 < 2; i2++) {
            int et = (w & 1) * 2 + i2;
            #pragma unroll
            for (int j = 0; j < 8; j++) {
                int rloc = nb * 16 + j + hi * 8;
                float dinv = 1.0f / smem[OFF_DROW + rloc];
                og[(size_t)(n0 + rloc) * D_DIM + et * 16 + l15] = o[i2][j] * dinv;
            }
        }
        __syncthreads();
    }
}

extern "C" void kernel_launch(void* const* d_in, const int* in_sizes, int n_in,
                              void* d_out, int out_size, void* d_ws, size_t ws_size,
                              hipStream_t stream) {
    (void)in_sizes; (void)n_in; (void)out_size; (void)d_ws; (void)ws_size;
    const float* q    = (const float*)d_in[0];
    const float* k    = (const float*)d_in[1];
    const float* v    = (const float*)d_in[2];
    const float* proj = (const float*)d_in[3];
    float* out = (float*)d_out;

    hipFuncSetAttribute((const void*)performer_attn_kernel,
                        hipFuncAttributeMaxDynamicSharedMemorySize, SMEM_BYTES);
    performer_attn_kernel<<<dim3(8 * 16), dim3(TPB), SMEM_BYTES, stream>>>(q, k, v, proj, out);
}